// uMPS_55654186222260
// MI455X (gfx1250) — compile-verified
//
#include <hip/hip_runtime.h>
#include <hip/hip_bf16.h>

typedef __attribute__((ext_vector_type(2))) float v2f;
typedef __attribute__((ext_vector_type(8))) float v8f;
typedef __attribute__((ext_vector_type(4))) unsigned int v4u;
typedef __attribute__((ext_vector_type(4))) int v4i;
typedef __attribute__((ext_vector_type(8))) int v8i;

#define NQ       16
#define LAYERS   3
#define BATCH    96
#define NPOS     (NQ - 1)              // 15 gate positions per layer
#define NGATES   (LAYERS * NPOS)       // 45
#define NSTATE   (1 << NQ)             // 65536 amplitudes (256 KB f32)
#define NGROUPS  (1 << (NQ - 2))       // 16384 4-amplitude gate groups
#define NWMMA    (NGROUPS / 16)        // 1024 wmma tiles per gate
#define NTHREADS 1024                  // 32 wave32 waves per workgroup
#define NWAVES   (NTHREADS / 32)
#define PWORDS   (NGATES * 16)         // 720 dwords of gate parameters

// One workgroup per batch element. Full state vector lives in the 320KB WGP LDS.
__global__ __launch_bounds__(NTHREADS) void umps_overlap_kernel(
    const float* __restrict__ X,       // (16, 96, 2)
    const float* __restrict__ label,   // (2, 96, 2)
    const float* __restrict__ params,  // (45, 2,2,2,2) == (45,16) row-major [k=l*2+m][c=r*2+s]
    float* __restrict__ out)           // (96)
{
    extern __shared__ float smem[];
    float* state  = smem;                      // NSTATE
    float* red    = smem + NSTATE;             // NTHREADS
    float* xn     = red + NTHREADS;            // 2*NQ
    float* pstage = xn + 2 * NQ;               // PWORDS (TDM-staged gate params)

    const int b    = blockIdx.x;
    const int tid  = threadIdx.x;
    const int lane = tid & 31;
    const int wave = tid >> 5;

    __builtin_prefetch(label, 0, 0);           // global_prefetch_b8

    // ---- TDM: DMA the whole gate table (720 dwords) global -> LDS, wave 0 issues ----
    if (wave == 0) {
        unsigned long long ga = (unsigned long long)(uintptr_t)params;
        unsigned int lds_off  = (unsigned int)(uintptr_t)pstage; // low 32 bits = LDS byte addr
        v4u g0;
        g0[0] = 1u;                                    // count=1, user desc, no gather
        g0[1] = lds_off;                               // lds_addr
        g0[2] = (unsigned int)(ga & 0xffffffffu);      // global_addr[31:0]
        g0[3] = (unsigned int)((ga >> 32) & 0x1ffffffu) | (2u << 30); // addr[56:32] | type=2
        v8i g1;
        g1[0] = (int)(2u << 16);                       // data_size=4B; mask/flags=0
        g1[1] = (int)(((unsigned)PWORDS & 0xffffu) << 16);   // tensor_dim0[15:0]
        g1[2] = (int)(((unsigned)PWORDS >> 16) | (1u << 16)); // tensor_dim0[31:16] | tensor_dim1=1
        g1[3] = (int)(((unsigned)PWORDS & 0xffffu) << 16);   // tile_dim0 = 720
        g1[4] = 1;                                     // tile_dim1=1, tile_dim2=0
        g1[5] = PWORDS;                                // tensor_dim0_stride[31:0]
        g1[6] = 0;                                     // stride hi, dim1_stride lo
        g1[7] = 0;
        v4i z4 = {};
#if defined(__clang_major__) && (__clang_major__ >= 23)
        v8i z8 = {};
        __builtin_amdgcn_tensor_load_to_lds(g0, g1, z4, z4, z8, 0);
#else
        __builtin_amdgcn_tensor_load_to_lds(g0, g1, z4, z4, 0);
#endif
    }

    // ---- normalize the 16 input 2-vectors for this batch element ----
    if (tid < NQ) {
        float x0 = X[(size_t)tid * BATCH * 2 + (size_t)b * 2 + 0];
        float x1 = X[(size_t)tid * BATCH * 2 + (size_t)b * 2 + 1];
        float r  = rsqrtf(x0 * x0 + x1 * x1);
        xn[2 * tid + 0] = x0 * r;
        xn[2 * tid + 1] = x1 * r;
    }
    __syncthreads();

    // ---- product state (overlaps the TDM transfer) ----
    for (int idx = tid; idx < NSTATE; idx += NTHREADS) {
        float p = 1.0f;
        #pragma unroll
        for (int q = 0; q < NQ; ++q) {
            int bit = (idx >> (NQ - 1 - q)) & 1;
            p *= xn[2 * q + bit];
        }
        state[idx] = p;
    }
    if (wave == 0) {
        __builtin_amdgcn_s_wait_tensorcnt(0);  // gate table resident in LDS
    }
    __syncthreads();

    // ---- staircase of 45 two-qubit gates via V_WMMA_F32_16X16X4_F32 ----
    // new[g][c] = sum_k old[g][k] * u4[k][c]   (u4 = 4x4 gate, k=l*2+m, c=r*2+s)
    // A (16x4): A[M][K] = u4[K][M&3]  (rows replicated mod 4 -> every D row valid)
    // B (4x16): B[K][N] = old[group N][K]
    // D (16x16): lane N, VGPR j -> new[group N][j&3]
    const int m4 = lane & 3;
    const int k0 = (lane < 16) ? 0 : 2;        // K pair held by this lane half
    for (int gi = 0; gi < NGATES; ++gi) {
        const int pos = gi % NPOS;             // qubits (pos, pos+1)
        const int s1  = 1 << (NQ - 2 - pos);   // stride of qubit pos+1
        const int s0  = s1 << 1;               // stride of qubit pos
        const float* u4 = pstage + gi * 16;    // LDS-resident gate

        v2f a;
        a.x = u4[(k0 + 0) * 4 + m4];
        a.y = u4[(k0 + 1) * 4 + m4];

        #pragma unroll 2
        for (int it = 0; it < NWMMA / NWAVES; ++it) {
            int w    = it * NWAVES + wave;
            int g    = (w << 4) | (lane & 15); // this lane's group (column N)
            int base = ((g >> (NQ - 2 - pos)) << (NQ - pos)) | (g & (s1 - 1));
            int addr = base + ((lane < 16) ? 0 : s0);

            v2f bm;
            bm.x = state[addr];                // K = k0     -> (l, m=0)
            bm.y = state[addr + s1];           // K = k0 + 1 -> (l, m=1)

            v8f c = {};
            c = __builtin_amdgcn_wmma_f32_16x16x4_f32(
                    false, a, false, bm, (short)0, c, false, false);

            // in-place: each lane writes exactly the 2 addresses it read
            state[addr]      = (lane < 16) ? c[0] : c[2];
            state[addr + s1] = (lane < 16) ? c[1] : c[3];
        }
        __syncthreads();
    }

    // ---- contract label leg (qubit 15, stride 1) with both labels, overlap ----
    const float l00 = label[0 * BATCH * 2 + b * 2 + 0];
    const float l01 = label[0 * BATCH * 2 + b * 2 + 1];
    const float l10 = label[1 * BATCH * 2 + b * 2 + 0];
    const float l11 = label[1 * BATCH * 2 + b * 2 + 1];
    float acc = 0.0f;
    for (int s = tid; s < NSTATE / 2; s += NTHREADS) {
        float e = state[2 * s + 0];
        float o = state[2 * s + 1];
        acc += (e * l00 + o * l01) * (e * l10 + o * l11);
    }
    red[tid] = acc;
    __syncthreads();
    for (int off = NTHREADS / 2; off > 0; off >>= 1) {
        if (tid < off) red[tid] += red[tid + off];
        __syncthreads();
    }
    if (tid == 0) out[b] = red[0];
}

extern "C" void kernel_launch(void* const* d_in, const int* in_sizes, int n_in,
                              void* d_out, int out_size, void* d_ws, size_t ws_size,
                              hipStream_t stream) {
    (void)in_sizes; (void)n_in; (void)out_size; (void)d_ws; (void)ws_size;
    const float* X      = (const float*)d_in[0];
    const float* label  = (const float*)d_in[1];
    const float* params = (const float*)d_in[2];
    float* out          = (float*)d_out;

    const size_t shmem =
        (size_t)(NSTATE + NTHREADS + 2 * NQ + PWORDS) * sizeof(float); // ~263 KB
    hipFuncSetAttribute(reinterpret_cast<const void*>(umps_overlap_kernel),
                        hipFuncAttributeMaxDynamicSharedMemorySize, (int)shmem);
    umps_overlap_kernel<<<BATCH, NTHREADS, shmem, stream>>>(X, label, params, out);
}